// GCN_86242943303861
// MI455X (gfx1250) — compile-verified
//
#include <hip/hip_runtime.h>

// Problem constants (match reference).
#define NN 100000   // nodes (6250 * 16 -> exact 16-row WMMA tiles)
#define DD 128      // feature dim
#define EE 1600000  // edges

typedef __attribute__((ext_vector_type(2))) float v2f;
typedef __attribute__((ext_vector_type(8))) float v8f;

// ---------------- degree / normalization ----------------

__global__ void k_init_deg(float* __restrict__ deg) {
  int i = blockIdx.x * blockDim.x + threadIdx.x;
  if (i < NN) deg[i] = 1.0f;  // self-loop included
}

__global__ void k_edge_deg(const long long* __restrict__ dst, float* __restrict__ deg) {
  int i = blockIdx.x * blockDim.x + threadIdx.x;
  if (i < EE) atomicAdd(&deg[(int)dst[i]], 1.0f);
}

__global__ void k_rsqrt_inplace(float* __restrict__ deg) {
  int i = blockIdx.x * blockDim.x + threadIdx.x;
  if (i < NN) deg[i] = rsqrtf(deg[i]);  // deg -> dis
}

// ---------------- dense GEMM via V_WMMA_F32_16X16X4_F32 ----------------
// out[M x 128] = A[M x 128] @ W[128 x 128] (+ bias, relu if fused).
// Block = 128 threads = 4 waves; each wave owns one 16x16 output tile.
// grid = (M/16, 128/(16*4)).

__global__ __launch_bounds__(128)
void k_gemm_wmma(const float* __restrict__ A, const float* __restrict__ W,
                 const float* __restrict__ bias, float* __restrict__ out,
                 int fuse_bias_relu) {
  const int lane = threadIdx.x & 31;
  const int wave = threadIdx.x >> 5;
  const int l15  = lane & 15;
  const int half = lane >> 4;  // 0: lanes 0-15, 1: lanes 16-31
  const int rowBase = blockIdx.x * 16;
  const int colBase = (blockIdx.y * 4 + wave) * 16;

  // A fragment source: row per lane (lanes 0-15 and 16-31 both map to rows 0-15).
  const float* __restrict__ arow = A + (size_t)(rowBase + l15) * DD;
  // B fragment source: column per lane.
  const float* __restrict__ bcol = W + colBase + l15;

  v8f c = {};
#pragma unroll
  for (int k = 0; k < DD; k += 4) {
    // 32-bit A 16x4 layout: VGPR0 = K=k (lanes 0-15) / K=k+2 (lanes 16-31),
    //                       VGPR1 = K=k+1            / K=k+3.
    v2f a, b;
    a.x = arow[k + 2 * half + 0];
    a.y = arow[k + 2 * half + 1];
    // B 4x16 mirrored layout (coalesced: lanes sweep 16 consecutive columns).
    b.x = bcol[(size_t)(k + 2 * half + 0) * DD];
    b.y = bcol[(size_t)(k + 2 * half + 1) * DD];
    // 8-arg form: (neg_a, A, neg_b, B, c_mod, C, reuse_a, reuse_b)
    c = __builtin_amdgcn_wmma_f32_16x16x4_f32(false, a, false, b, (short)0, c,
                                              false, false);
  }

  float bb = 0.0f;
  if (fuse_bias_relu) bb = bias[colBase + l15];
  // C/D layout: VGPR r -> row r (lanes 0-15) or row r+8 (lanes 16-31), N = lane%16.
#pragma unroll
  for (int r = 0; r < 8; ++r) {
    int row = rowBase + r + 8 * half;
    float v = c[r];
    if (fuse_bias_relu) v = fmaxf(v + bb, 0.0f);
    out[(size_t)row * DD + colBase + l15] = v;
  }
}

// ---------------- GCN aggregation ----------------

// agg[i,:] = dis[i]^2 * xw[i,:]  (self-loop term; also re-initializes ws/out).
__global__ void k_selfloop_init(const float* __restrict__ xw,
                                const float* __restrict__ dis,
                                float* __restrict__ agg) {
  int i = blockIdx.x * blockDim.x + threadIdx.x;
  if (i < NN * DD) {
    int row = i >> 7;  // /128
    float d = dis[row];
    agg[i] = d * d * xw[i];
  }
}

// One wave per edge, 4 features per lane: float4 gather from xw[src], scaled
// fp32 atomic adds into agg[dst]. xw/agg (51 MB each) live in the 192 MB L2,
// so the random traffic mostly stays on-chip.
__global__ __launch_bounds__(256)
void k_scatter(const long long* __restrict__ src, const long long* __restrict__ dst,
               const float* __restrict__ dis, const float* __restrict__ xw,
               float* __restrict__ agg) {
  int tid  = blockIdx.x * blockDim.x + threadIdx.x;
  int e    = tid >> 5;
  int lane = tid & 31;
  if (e >= EE) return;
  int s = (int)src[e];
  int d = (int)dst[e];
  float coef = dis[s] * dis[d];
  const float4* xs = (const float4*)(xw + (size_t)s * DD);
  float4 v = xs[lane];
  float* ad = agg + (size_t)d * DD + lane * 4;
  atomicAdd(ad + 0, v.x * coef);
  atomicAdd(ad + 1, v.y * coef);
  atomicAdd(ad + 2, v.z * coef);
  atomicAdd(ad + 3, v.w * coef);
}

__global__ void k_bias_act(float* __restrict__ h, const float* __restrict__ bias,
                           int do_relu) {
  int i = blockIdx.x * blockDim.x + threadIdx.x;
  if (i < NN * DD) {
    float v = h[i] + bias[i & (DD - 1)];
    h[i] = do_relu ? fmaxf(v, 0.0f) : v;
  }
}

// ---------------- launch ----------------

extern "C" void kernel_launch(void* const* d_in, const int* in_sizes, int n_in,
                              void* d_out, int out_size, void* d_ws, size_t ws_size,
                              hipStream_t stream) {
  (void)in_sizes; (void)n_in; (void)out_size; (void)ws_size;
  const float*     x  = (const float*)d_in[0];
  const long long* ei = (const long long*)d_in[1];  // [2, E] int64
  const float* W1 = (const float*)d_in[2]; const float* b1 = (const float*)d_in[3];
  const float* W2 = (const float*)d_in[4]; const float* b2 = (const float*)d_in[5];
  const float* Wv = (const float*)d_in[6]; const float* bv = (const float*)d_in[7];
  const float* Wt = (const float*)d_in[8]; const float* bt = (const float*)d_in[9];

  const long long* srcI = ei;
  const long long* dstI = ei + EE;

  float* out_h = (float*)d_out;                       // h         [N, D]
  float* out_v = out_h + (size_t)NN * DD;             // x_vision  [N, D]
  float* out_t = out_v + (size_t)NN * DD;             // x_text    [N, D]

  float* ws   = (float*)d_ws;
  float* dis  = ws;                                   // N floats (deg -> dis)
  float* bufA = ws + NN;                              // xw  [N, D]
  float* bufB = bufA + (size_t)NN * DD;               // agg1 / h1 [N, D]

  const int T = 256;
  const int gN   = (NN + T - 1) / T;
  const int gE   = (EE + T - 1) / T;
  const int gND  = (NN * DD + T - 1) / T;
  const int gEW  = (EE * 32 + T - 1) / T;             // wave-per-edge scatter
  const dim3 gemmGrid(NN / 16, DD / (16 * 4));        // (6250, 2)

  // normalization: deg = 1 + indegree; dis = rsqrt(deg)
  k_init_deg<<<gN, T, 0, stream>>>(dis);
  k_edge_deg<<<gE, T, 0, stream>>>(dstI, dis);
  k_rsqrt_inplace<<<gN, T, 0, stream>>>(dis);

  // ---- layer 1: h1 = relu(GCN(x, W1) + b1) ----
  k_gemm_wmma<<<gemmGrid, 128, 0, stream>>>(x, W1, nullptr, bufA, 0);       // xw1
  k_selfloop_init<<<gND, T, 0, stream>>>(bufA, dis, bufB);                  // agg1
  k_scatter<<<gEW, T, 0, stream>>>(srcI, dstI, dis, bufA, bufB);
  k_bias_act<<<gND, T, 0, stream>>>(bufB, b1, 1);                           // h1

  // ---- layer 2: h = GCN(h1, W2) + b2 (no relu), written to d_out ----
  k_gemm_wmma<<<gemmGrid, 128, 0, stream>>>(bufB, W2, nullptr, bufA, 0);    // xw2
  k_selfloop_init<<<gND, T, 0, stream>>>(bufA, dis, out_h);                 // agg2
  k_scatter<<<gEW, T, 0, stream>>>(srcI, dstI, dis, bufA, out_h);
  k_bias_act<<<gND, T, 0, stream>>>(out_h, b2, 0);                          // h

  // ---- heads: fused bias + relu GEMMs ----
  k_gemm_wmma<<<gemmGrid, 128, 0, stream>>>(out_h, Wv, bv, out_v, 1);
  k_gemm_wmma<<<gemmGrid, 128, 0, stream>>>(out_h, Wt, bt, out_t, 1);
}